// Loss_9483287790299
// MI455X (gfx1250) — compile-verified
//
#include <hip/hip_runtime.h>

typedef float v2f __attribute__((ext_vector_type(2)));
typedef float v8f __attribute__((ext_vector_type(8)));

#define NROWS 8192
#define DDIM  256
#define NTILE (NROWS / 16)   // 512 column tiles
#define MARGIN_F 0.2f

// diag[i] = <im_i, s_i>; one wave32 per row.
__global__ void diag_kernel(const float* __restrict__ im,
                            const float* __restrict__ s,
                            float* __restrict__ diag) {
    const int lane = threadIdx.x & 31;
    const int wave = threadIdx.x >> 5;
    const int row  = blockIdx.x * 8 + wave;
    const float* a = im + (size_t)row * DDIM;
    const float* b = s  + (size_t)row * DDIM;
    float acc = 0.f;
    #pragma unroll
    for (int k = lane; k < DDIM; k += 32) acc += a[k] * b[k];
    #pragma unroll
    for (int off = 16; off; off >>= 1) acc += __shfl_xor(acc, off, 32);
    if (lane == 0) diag[row] = acc;
}

// Load one 16x256 B tile as 64 K-step fragments (v2f per lane per step).
__device__ __forceinline__ void load_btile(v2f (&buf)[64], const float* __restrict__ B,
                                           int j0, int ml, int kb) {
    const float* brow = B + (size_t)(j0 + ml) * DDIM + kb;
    #pragma unroll
    for (int k = 0; k < 64; ++k)
        buf[k] = *(const v2f*)(brow + 4 * k);
}

// 64-step WMMA chain (two independent accumulators) + rank/hinge epilogue.
__device__ __forceinline__ void compute_tile(const v2f (&buf)[64], const v2f (&afrag)[64],
                                             int i0, int j0, int ml, int half,
                                             const float (&dg)[8],
                                             float (&cnt)[8], float (&mx)[8]) {
    v8f c0 = {}, c1 = {};
    #pragma unroll
    for (int k = 0; k < 64; k += 2) {
        c0 = __builtin_amdgcn_wmma_f32_16x16x4_f32(
                 false, afrag[k],     false, buf[k],     (short)0, c0, false, false);
        c1 = __builtin_amdgcn_wmma_f32_16x16x4_f32(
                 false, afrag[k + 1], false, buf[k + 1], (short)0, c1, false, false);
    }
    const v8f c = c0 + c1;
    const int jg = j0 + ml;  // global column of this lane's elements
    #pragma unroll
    for (int r = 0; r < 8; ++r) {
        const int   ig = i0 + r + 8 * half;   // C/D layout: M = r + 8*half
        const float sc = c[r];
        cnt[r] += (sc < dg[r]) ? 1.f : 0.f;
        float cost = MARGIN_F + sc - dg[r];
        cost = (ig == jg) ? 0.f : cost;       // zero the positive pair
        cost = fmaxf(cost, 0.f);
        mx[r] = fmaxf(mx[r], cost);
    }
}

// Fused GEMM + epilogue: for each row i of A (vs all rows j of B):
//   out_cnt[i] = #{ j : <A_i,B_j> < diag[i] }
//   out_mx[i]  = max_{j != i} max(MARGIN + <A_i,B_j> - diag[i], 0)
// One block per 16-row tile; 4 waves split the 512 column tiles.
// B tiles are register double-buffered so all 64 loads of the next tile are
// in flight while the current tile's WMMA chain executes.
__global__ void __launch_bounds__(128)
stats_kernel(const float* __restrict__ A, const float* __restrict__ B,
             const float* __restrict__ diag,
             float* __restrict__ out_cnt, float* __restrict__ out_mx) {
    const int lane = threadIdx.x & 31;
    const int wave = threadIdx.x >> 5;
    const int i0   = blockIdx.x * 16;
    const int half = lane >> 4;      // 0: lanes 0-15, 1: lanes 16-31
    const int ml   = lane & 15;      // A-frag row / B-frag column
    const int kb   = half * 2;       // K sub-offset within a K=4 step

    // Preload the whole 16x256 A tile as 64 K-step fragments (kept in VGPRs).
    v2f afrag[64];
    {
        const float* arow = A + (size_t)(i0 + ml) * DDIM + kb;
        #pragma unroll
        for (int k = 0; k < 64; ++k)
            afrag[k] = *(const v2f*)(arow + 4 * k);
    }

    // diag values for the 8 output rows this lane owns.
    float dg[8];
    #pragma unroll
    for (int r = 0; r < 8; ++r) dg[r] = diag[i0 + r + 8 * half];

    float cnt[8], mx[8];
    #pragma unroll
    for (int r = 0; r < 8; ++r) { cnt[r] = 0.f; mx[r] = 0.f; }

    // Each wave owns tiles {wave, wave+4, wave+8, ...}: exactly 128 tiles.
    v2f b0[64], b1[64];
    load_btile(b0, B, wave * 16, ml, kb);
    #pragma unroll 1
    for (int idx = 0; idx < 128; idx += 2) {
        const int jt0 = wave + (idx + 0) * 4;
        const int jt1 = wave + (idx + 1) * 4;           // always < NTILE
        int jt2 = wave + (idx + 2) * 4;
        jt2 = (jt2 < NTILE) ? jt2 : wave;               // clamp final prefetch (unused)

        load_btile(b1, B, jt1 * 16, ml, kb);            // prefetch tile idx+1
        compute_tile(b0, afrag, i0, jt0 * 16, ml, half, dg, cnt, mx);
        load_btile(b0, B, jt2 * 16, ml, kb);            // prefetch tile idx+2
        compute_tile(b1, afrag, i0, jt1 * 16, ml, half, dg, cnt, mx);
    }

    // Reduce across the 16 lanes of each half (xor masks 1..8 stay in-half on wave32).
    #pragma unroll
    for (int off = 1; off <= 8; off <<= 1) {
        #pragma unroll
        for (int r = 0; r < 8; ++r) {
            cnt[r] += __shfl_xor(cnt[r], off, 32);
            mx[r]   = fmaxf(mx[r], __shfl_xor(mx[r], off, 32));
        }
    }

    __shared__ float s_cnt[4][16];
    __shared__ float s_mx[4][16];
    if (ml == 0) {
        #pragma unroll
        for (int r = 0; r < 8; ++r) {
            s_cnt[wave][r + 8 * half] = cnt[r];
            s_mx [wave][r + 8 * half] = mx[r];
        }
    }
    __syncthreads();
    if (threadIdx.x < 16) {
        float tc = 0.f, tm = 0.f;
        #pragma unroll
        for (int w = 0; w < 4; ++w) {
            tc += s_cnt[w][threadIdx.x];
            tm  = fmaxf(tm, s_mx[w][threadIdx.x]);
        }
        out_cnt[i0 + threadIdx.x] = tc;
        out_mx [i0 + threadIdx.x] = tm;
    }
}

// loss = sum_i mx1[i]/(cnt1[i]+1) + mx2[i]/(cnt2[i]+1)
__global__ void final_kernel(const float* __restrict__ cnt1, const float* __restrict__ mx1,
                             const float* __restrict__ cnt2, const float* __restrict__ mx2,
                             float* __restrict__ out) {
    __shared__ float red[256];
    float acc = 0.f;
    for (int i = threadIdx.x; i < NROWS; i += 256)
        acc += mx1[i] / (cnt1[i] + 1.f) + mx2[i] / (cnt2[i] + 1.f);
    red[threadIdx.x] = acc;
    __syncthreads();
    for (int off = 128; off; off >>= 1) {
        if ((int)threadIdx.x < off) red[threadIdx.x] += red[threadIdx.x + off];
        __syncthreads();
    }
    if (threadIdx.x == 0) out[0] = red[0];
}

extern "C" void kernel_launch(void* const* d_in, const int* in_sizes, int n_in,
                              void* d_out, int out_size, void* d_ws, size_t ws_size,
                              hipStream_t stream) {
    const float* im = (const float*)d_in[0];
    const float* s  = (const float*)d_in[1];
    float* ws   = (float*)d_ws;
    float* diag = ws;                 // [N]
    float* cnt1 = ws + 1 * NROWS;     // [N]
    float* mx1  = ws + 2 * NROWS;     // [N]
    float* cnt2 = ws + 3 * NROWS;     // [N]
    float* mx2  = ws + 4 * NROWS;     // [N]

    diag_kernel <<<NROWS / 8, 256, 0, stream>>>(im, s, diag);
    stats_kernel<<<NROWS / 16, 128, 0, stream>>>(im, s, diag, cnt1, mx1); // row stats
    stats_kernel<<<NROWS / 16, 128, 0, stream>>>(s, im, diag, cnt2, mx2); // col stats
    final_kernel<<<1, 256, 0, stream>>>(cnt1, mx1, cnt2, mx2, (float*)d_out);
}